// Embedding_32452772889204
// MI455X (gfx1250) — compile-verified
//
#include <hip/hip_runtime.h>
#include <stdint.h>

typedef __attribute__((ext_vector_type(4))) unsigned int u32x4;
typedef __attribute__((ext_vector_type(8))) unsigned int u32x8;
typedef __attribute__((ext_vector_type(4))) int          i32x4;
typedef __attribute__((ext_vector_type(4))) float        f32x4;

#define EMB_D        2048
#define ROWS_PER_WG  8     // TDM gather mode, 32-bit indices: max 8 rows/descriptor

__global__ __launch_bounds__(256) void embed_tdm_gather_kernel(
    const int*   __restrict__ x,        // [nrows] row indices
    const int*   __restrict__ weight,   // [vocab, EMB_D] int8 values stored as int32
    const float* __restrict__ wscale,   // [vocab] per-row scale
    float*       __restrict__ out,      // [nrows, EMB_D]
    int nrows, int vocab)
{
    __shared__ int lds_buf[ROWS_PER_WG * EMB_D];   // 64 KB staging tile

    const int lane = threadIdx.x & 31;
    const int wave = threadIdx.x >> 5;             // wave32
    const int base = blockIdx.x * ROWS_PER_WG;

    // Per-wave row index + scale. threadIdx.x>>5 is wave-uniform but LLVM's
    // divergence analysis can't see that, so launder via readfirstlane to get
    // SGPR addresses -> scalar (constant-cache) loads instead of vector loads.
    const int   row    = base + wave;
    const int   rowc   = __builtin_amdgcn_readfirstlane(row < nrows ? row : (nrows - 1));
    const int   my_idx = __builtin_amdgcn_readfirstlane(x[rowc]);   // s_load
    const float s      = wscale[my_idx];                            // s_load

    if (wave == 0) {
        // ---- Build Tensor DMA Descriptor (D#) for an 8-row gather ----
        unsigned idx[ROWS_PER_WG];
        if (base + ROWS_PER_WG <= nrows) {
            // Common case: one 32B-aligned s_load_b256 of the 8 indices
            // (base*4 bytes = blockIdx.x*32 -> 32-byte aligned).
            const u32x8 iv = *(const u32x8*)(x + base);
            #pragma unroll
            for (int r = 0; r < ROWS_PER_WG; ++r) idx[r] = iv[r];
        } else {
            // Ragged tail: clamped scalar loads.
            #pragma unroll
            for (int r = 0; r < ROWS_PER_WG; ++r) {
                int rr = base + r;
                idx[r] = (unsigned)x[rr < nrows ? rr : (nrows - 1)];
            }
        }

        unsigned lds_addr = (unsigned)(uintptr_t)(&lds_buf[0]);
        unsigned long long gaddr = (unsigned long long)(uintptr_t)weight;

        // Group 0 (128b): count=1 | gather_index_size=1 (32-bit) | gather_mode=1,
        // lds_addr[63:32], global_addr[120:64], type=2 ("image") in [127:126].
        u32x4 g0;
        g0[0] = 0xC0000001u;
        g0[1] = lds_addr;
        g0[2] = (unsigned)(gaddr & 0xFFFFFFFFu);
        g0[3] = (unsigned)((gaddr >> 32) & 0x01FFFFFFu) | (2u << 30);

        // Group 1 (256b): workgroup_mask=0 (not in cluster), data_size=2 (4B),
        // tensor_dim0 = EMB_D, tensor_dim1 = vocab, tile_dim0 = EMB_D,
        // tile_dim1 = ROWS_PER_WG (8), tensor_dim0_stride = EMB_D.
        u32x8 g1;
        g1[0] = (2u << 16);                                            // data_size=4B
        g1[1] = ((unsigned)EMB_D & 0xFFFFu) << 16;                     // tensor_dim0[15:0]
        g1[2] = ((unsigned)vocab & 0xFFFFu) << 16;                     // dim0 hi=0, tensor_dim1[15:0]
        g1[3] = (((unsigned)vocab >> 16) & 0xFFFFu)                    // tensor_dim1[31:16]
              | (((unsigned)EMB_D & 0xFFFFu) << 16);                   // tile_dim0
        g1[4] = (unsigned)ROWS_PER_WG;                                 // tile_dim1 = #indices
        g1[5] = (unsigned)EMB_D;                                       // tensor_dim0_stride lo32
        g1[6] = 0u;
        g1[7] = 0u;

        // Groups 2-3: eight 32-bit row indices.
        u32x4 g2; g2[0] = idx[0]; g2[1] = idx[1]; g2[2] = idx[2]; g2[3] = idx[3];
        u32x4 g3; g3[0] = idx[4]; g3[1] = idx[5]; g3[2] = idx[6]; g3[3] = idx[7];

        asm volatile("tensor_load_to_lds %0, %1, %2, %3"
                     :: "s"(g0), "s"(g1), "s"(g2), "s"(g3)
                     : "memory");
        __builtin_amdgcn_s_wait_tensorcnt(0);   // TDM done -> LDS tile valid
    }
    __syncthreads();                            // release other 7 waves

    // ---- Dequantize: wave w handles gathered row w ----
    if (row < nrows) {
        const int* lrow = &lds_buf[wave * EMB_D];
        float*     orow = out + (size_t)row * EMB_D;

        #pragma unroll
        for (int it = 0; it < EMB_D / (32 * 4); ++it) { // 16 iterations
            const int e = (it * 32 + lane) * 4;
            i32x4 q = *(const i32x4*)(lrow + e);        // ds_load_b128
            f32x4 f;
            f[0] = (float)q[0] * s;
            f[1] = (float)q[1] * s;
            f[2] = (float)q[2] * s;
            f[3] = (float)q[3] * s;
            __builtin_nontemporal_store(f, (f32x4*)(orow + e)); // streaming b128 store
        }
    }
}

extern "C" void kernel_launch(void* const* d_in, const int* in_sizes, int n_in,
                              void* d_out, int out_size, void* d_ws, size_t ws_size,
                              hipStream_t stream) {
    const int*   x      = (const int*)d_in[0];    // indices (int per harness contract)
    const int*   weight = (const int*)d_in[1];    // [V, D] int32 holding int8 values
    const float* wscale = (const float*)d_in[2];  // [V]
    float*       out    = (float*)d_out;          // [B*S, D] float32

    const int nrows = in_sizes[0];                // B*S = 32768
    const int vocab = in_sizes[2];                // V = 128000

    const int grid = (nrows + ROWS_PER_WG - 1) / ROWS_PER_WG;
    embed_tdm_gather_kernel<<<grid, 256, 0, stream>>>(x, weight, wscale, out,
                                                      nrows, vocab);
}